// FootAndBall_65515431133906
// MI455X (gfx1250) — compile-verified
//
#include <hip/hip_runtime.h>
#include <stdint.h>

// ---------------- problem constants (from reference) ----------------
#define B_N   16
#define H_N   540
#define W_N   960
#define HW_N  (H_N * W_N)
#define K_DET 100
// DOWNSCALE=4 -> center offset (4-1)/2 = 1.5 ; half box = 10

// ---------------- pass-1 tiling ----------------
#define TILE_H 32
#define TILE_W 64          // 64 floats per LDS row; 16 chunks of 16B
#define INT_H  30          // interior rows  (1-row halo top/bottom)
#define INT_W  56          // interior cols  (4-col halo left/right -> 16B aligned chunks)
#define P1_THREADS 512

// ---------------- pass-2 selection ----------------
#define H1_BITS 13
#define H1_BINS (1 << H1_BITS)     // 8192 bins on top 13 key bits
#define COMP_CAP 512
#define CAND_CAP_MAX 65536

// ---------- CDNA5 async global->LDS path (probe via __has_builtin) ----------
typedef int v4i_ __attribute__((ext_vector_type(4)));
#define AS1_ __attribute__((address_space(1)))
#define AS3_ __attribute__((address_space(3)))

#if __has_builtin(__builtin_amdgcn_global_load_async_to_lds_b128)
#define HAVE_ASYNC_LDS 1
#define ASYNC_CP128(dst_lds, src_glb)                                          \
    __builtin_amdgcn_global_load_async_to_lds_b128(                            \
        (AS1_ v4i_*)(AS1_ void*)(void*)(src_glb),                              \
        (AS3_ v4i_*)(AS3_ void*)(dst_lds), 0, 0)
#endif

#if __has_builtin(__builtin_amdgcn_s_wait_asynccnt)
#define ASYNC_WAIT0() __builtin_amdgcn_s_wait_asynccnt(0)
#else
#define ASYNC_WAIT0() asm volatile("s_wait_asynccnt 0" ::: "memory")
#endif

// Pass 1: async-stage raw channels to LDS, sigmoid + 3x3 NMS, candidate append
__global__ __launch_bounds__(P1_THREADS)
void fb_pass1(const float* __restrict__ in,
              unsigned int* __restrict__ counters,
              unsigned long long* __restrict__ cand,
              int candCap)
{
    __shared__ float raw0[TILE_H * TILE_W];
    __shared__ float raw1[TILE_H * TILE_W];
    __shared__ float pt  [TILE_H * TILE_W];

    const int tid = threadIdx.x;
    const int tX = blockIdx.x, tY = blockIdx.y, b = blockIdx.z;
    const int gc0 = tX * INT_W - 4;   // global col of LDS col 0 (multiple of 4)
    const int gr0 = tY * INT_H - 1;   // global row of LDS row 0

    const float* base0 = in + (size_t)b * 2 * HW_N; // channel 0
    const float* base1 = base0 + HW_N;              // channel 1 (ball)

    // ---- stage one 16B chunk per thread per channel into LDS ----
    {
        int row = tid >> 4;            // 0..31
        int cg  = (tid & 15) << 2;     // LDS col of chunk start (mult of 4)
        int gr = gr0 + row; gr = gr < 0 ? 0 : (gr > H_N - 1 ? H_N - 1 : gr);
        int gc = gc0 + cg;  gc = gc < 0 ? 0 : (gc > W_N - 4 ? W_N - 4 : gc);
        // chunks are 4-aligned in global cols, so clamping only relocates
        // chunks whose 4 columns are ALL out of image (halo); no corruption.
        const float* g0 = base0 + (size_t)gr * W_N + gc;
        const float* g1 = base1 + (size_t)gr * W_N + gc;
        float* l0 = &raw0[row * TILE_W + cg];
        float* l1 = &raw1[row * TILE_W + cg];
#ifdef HAVE_ASYNC_LDS
        ASYNC_CP128(l0, g0);
        ASYNC_CP128(l1, g1);
        ASYNC_WAIT0();
#else
        *(float4*)l0 = *(const float4*)g0;
        *(float4*)l1 = *(const float4*)g1;
#endif
    }
    __syncthreads();

    // ---- p = sigmoid(x1 - x0); OOB coords marked -1 ----
    for (int pos = tid; pos < TILE_H * TILE_W; pos += P1_THREADS) {
        int row = pos >> 6, col = pos & 63;
        int gr = gr0 + row, gc = gc0 + col;
        float p;
        if (gr >= 0 && gr < H_N && gc >= 0 && gc < W_N) {
            float z = raw1[pos] - raw0[pos];
            p = 1.0f / (1.0f + __expf(-z));
        } else {
            p = -1.0f;            // acts like -inf padding of reduce_window
        }
        pt[pos] = p;
    }
    __syncthreads();

    // ---- 3x3 NMS on interior; append surviving maxima ----
    for (int pos = tid; pos < TILE_H * TILE_W; pos += P1_THREADS) {
        int row = pos >> 6, col = pos & 63;
        if (row < 1 || row > INT_H) continue;
        if (col < 4 || col >= 4 + INT_W) continue;
        int gr = gr0 + row, gc = gc0 + col;
        if (gr >= H_N || gc >= W_N) continue;
        float c = pt[pos];
        float m = c;
        m = fmaxf(m, pt[pos - 1]);             m = fmaxf(m, pt[pos + 1]);
        m = fmaxf(m, pt[pos - TILE_W - 1]);    m = fmaxf(m, pt[pos - TILE_W]);
        m = fmaxf(m, pt[pos - TILE_W + 1]);    m = fmaxf(m, pt[pos + TILE_W - 1]);
        m = fmaxf(m, pt[pos + TILE_W]);        m = fmaxf(m, pt[pos + TILE_W + 1]);
        if (c >= m) {  // local maximum (ties kept, as in reference)
            unsigned int pix = (unsigned int)(gr * W_N + gc);
            unsigned long long key =
                ((unsigned long long)__float_as_uint(c) << 32) |
                (unsigned long long)(0xFFFFFFFFu - pix);  // value desc, index asc
            unsigned int slot = atomicAdd(&counters[b], 1u);
            if ((int)slot < candCap)
                cand[(size_t)b * (size_t)candCap + slot] = key;
        }
    }
}

__global__ __launch_bounds__(64) void fb_zero_counters(unsigned int* c) {
    if (threadIdx.x < B_N) c[threadIdx.x] = 0u;
}

__device__ inline unsigned long long shflxor_u64(unsigned long long v, int mask) {
    unsigned int lo = (unsigned int)v, hi = (unsigned int)(v >> 32);
    lo = __shfl_xor(lo, mask, 32);
    hi = __shfl_xor(hi, mask, 32);
    return ((unsigned long long)hi << 32) | lo;
}

// Pass 2: exact per-batch top-100 via 2-level radix select + wave extraction
__global__ __launch_bounds__(1024)
void fb_pass2(const unsigned int* __restrict__ counters,
              const unsigned long long* __restrict__ cand,
              float* __restrict__ out, int candCap)
{
    __shared__ unsigned int hist1[H1_BINS];      // 32 KB
    __shared__ unsigned int groupsum[1024];      // 4 KB
    __shared__ unsigned int sub[64];
    __shared__ unsigned long long comp[COMP_CAP];// 4 KB
    __shared__ unsigned int s_T1, s_above1, s_T19, s_nsel;

    const int tid = threadIdx.x;
    const int b = blockIdx.x;
    const unsigned long long* cb = cand + (size_t)b * (size_t)candCap;
    int count = (int)counters[b];
    if (count > candCap) count = candCap;

    for (int i = tid; i < H1_BINS; i += 1024) hist1[i] = 0u;
    if (tid < 64) sub[tid] = 0u;
    if (tid == 0) s_nsel = 0u;
    __syncthreads();

    // level-1 histogram on top 13 key bits
    for (int i = tid; i < count; i += 1024)
        atomicAdd(&hist1[(unsigned int)(cb[i] >> 51)], 1u);
    __syncthreads();

    {   // per-thread group sums (8 bins each)
        unsigned int s = 0; int base = tid << 3;
        #pragma unroll
        for (int j = 0; j < 8; ++j) s += hist1[base + j];
        groupsum[tid] = s;
    }
    __syncthreads();

    if (tid == 0) {
        unsigned int cum = 0; int g;
        for (g = 1023; g > 0; --g) {
            if (cum + groupsum[g] >= K_DET) break;
            cum += groupsum[g];
        }
        int base = g << 3, bin;
        for (bin = base + 7; bin > base; --bin) {
            if (cum + hist1[bin] >= K_DET) break;
            cum += hist1[bin];
        }
        s_T1 = (unsigned int)bin;   // threshold bucket (level 1)
        s_above1 = cum;             // strictly above bucket (< K_DET)
    }
    __syncthreads();
    unsigned int T1 = s_T1;

    // level-2: 64 sub-bins (key bits 50..45) within bucket T1
    for (int i = tid; i < count; i += 1024) {
        unsigned long long k = cb[i];
        if ((unsigned int)(k >> 51) == T1)
            atomicAdd(&sub[(unsigned int)(k >> 45) & 63u], 1u);
    }
    __syncthreads();

    if (tid == 0) {
        unsigned int K2 = K_DET - s_above1;
        unsigned int cum = 0; int sb;
        for (sb = 63; sb > 0; --sb) {
            if (cum + sub[sb] >= K2) break;
            cum += sub[sb];
        }
        s_T19 = (T1 << 6) | (unsigned int)sb;  // 19-bit threshold prefix
    }
    __syncthreads();
    unsigned int T19 = s_T19;

    // compact survivors (top-19-bit prefix >= threshold) into LDS
    for (int i = tid; i < count; i += 1024) {
        unsigned long long k = cb[i];
        if ((unsigned int)(k >> 45) >= T19) {
            unsigned int pos = atomicAdd(&s_nsel, 1u);
            if (pos < COMP_CAP) comp[pos] = k;
        }
    }
    __syncthreads();
    int nsel = (int)s_nsel; if (nsel > COMP_CAP) nsel = COMP_CAP;

    // single wave: 100 exact max extractions, no barriers needed
    if (tid < 32) {
        float* ob = out + (size_t)b * K_DET * 5;
        for (int k = 0; k < K_DET; ++k) {
            unsigned long long lbest = 0ull; int larg = -1;
            for (int i = tid; i < nsel; i += 32) {
                unsigned long long v = comp[i];
                if (v > lbest) { lbest = v; larg = i; }
            }
            unsigned long long best = lbest;
            #pragma unroll
            for (int off = 16; off > 0; off >>= 1) {
                unsigned long long o = shflxor_u64(best, off);
                if (o > best) best = o;
            }
            if (best != 0ull && lbest == best) comp[larg] = 0ull; // keys unique
            if (tid == 0) {
                float val; unsigned int pix;
                if (best != 0ull) {
                    val = __uint_as_float((unsigned int)(best >> 32));
                    pix = 0xFFFFFFFFu - (unsigned int)(best & 0xFFFFFFFFull);
                } else { val = 0.0f; pix = 0u; }
                float xc = (float)(pix % W_N) * 4.0f + 1.5f;
                float yc = (float)(pix / W_N) * 4.0f + 1.5f;
                ob[k * 5 + 0] = xc - 10.0f;
                ob[k * 5 + 1] = yc - 10.0f;
                ob[k * 5 + 2] = xc + 10.0f;
                ob[k * 5 + 3] = yc + 10.0f;
                ob[k * 5 + 4] = val;
            }
        }
    }
}

extern "C" void kernel_launch(void* const* d_in, const int* in_sizes, int n_in,
                              void* d_out, int out_size, void* d_ws, size_t ws_size,
                              hipStream_t stream) {
    (void)in_sizes; (void)n_in; (void)out_size;
    const float* in = (const float*)d_in[0];
    float* out = (float*)d_out;

    unsigned int* counters = (unsigned int*)d_ws;
    unsigned long long* cand = (unsigned long long*)((char*)d_ws + 4096);
    size_t perBatch = ws_size > 4096 ? (ws_size - 4096) / ((size_t)B_N * 8ull) : 1;
    int candCap = (int)(perBatch < (size_t)CAND_CAP_MAX ? perBatch : (size_t)CAND_CAP_MAX);
    if (candCap < 1) candCap = 1;

    fb_zero_counters<<<1, 64, 0, stream>>>(counters);

    dim3 g1((W_N + INT_W - 1) / INT_W,   // 18 col tiles
            (H_N + INT_H - 1) / INT_H,   // 18 row tiles
            B_N);
    fb_pass1<<<g1, P1_THREADS, 0, stream>>>(in, counters, cand, candCap);

    fb_pass2<<<B_N, 1024, 0, stream>>>(counters, cand, out, candCap);
}